// NetVLADLike_32770600468821
// MI455X (gfx1250) — compile-verified
//
#include <hip/hip_runtime.h>
#include <hip/hip_bf16.h>
#include <math.h>

typedef float v2f __attribute__((ext_vector_type(2)));
typedef float v4f __attribute__((ext_vector_type(4)));
typedef float v8f __attribute__((ext_vector_type(8)));

#define BB 32
#define LL 2048
#define DD 512
#define KK 512
#define LAMBDA 200.0f          // 2 * CONC; per-column bias cancels in softmax over L
#define LT 32                  // rows of `direction` staged in LDS per chunk (2 wmma row tiles)
#define LDS_STRIDE 516         // 512 + 4: keeps float4 16B alignment; row r -> bank 4r (conflict-free)

// ---------------------------------------------------------------------------
// scale[b,d] = mean_l x[b,l,d]^2 ; rscale = 1/max(scale, eps)
__global__ void k_scale(const float* __restrict__ x, float* __restrict__ scale,
                        float* __restrict__ rscale) {
  int idx = blockIdx.x * blockDim.x + threadIdx.x;      // 0..B*D-1
  int b = idx >> 9, d = idx & (DD - 1);
  const float* xp = x + (size_t)b * LL * DD + d;
  float acc = 0.f;
  for (int l = 0; l < LL; ++l) {
    float v = xp[(size_t)l * DD];
    acc = fmaf(v, v, acc);
  }
  float s = acc * (1.0f / (float)LL);
  scale[idx]  = s;
  rscale[idx] = 1.0f / fmaxf(s, 1e-12f);
}

// csum[d] = sum_k centroids[k,d]   (since sum_l a[b,l,k] == 1 for every k)
__global__ void k_csum(const float* __restrict__ cent, float* __restrict__ csum) {
  int d = blockIdx.x * blockDim.x + threadIdx.x;
  if (d >= DD) return;
  float s = 0.f;
  for (int k = 0; k < KK; ++k) s += cent[(size_t)k * DD + d];
  csum[d] = s;
}

// ---------------------------------------------------------------------------
// Stage one 32 x 512 chunk of direction = x * rscale into LDS with b128 traffic.
__device__ inline void fill_dir(float* sdir, const float* srs,
                                const float* __restrict__ x, int b, int l0) {
  const float* xp = x + ((size_t)b * LL + l0) * DD;
  for (int i = threadIdx.x; i < LT * (DD / 4); i += 256) {
    int row = i >> 7;                   // DD/4 = 128 float4 per row
    int d4  = (i & 127) * 4;
    v4f v = *(const v4f*)(xp + (size_t)row * DD + d4);   // global_load_b128
    v4f r = *(const v4f*)(srs + d4);
    v *= r;
    *(v4f*)(sdir + row * LDS_STRIDE + d4) = v;           // ds_store_b128 (16B aligned)
  }
}

// One wave: two 16x16 tiles (rows 0-15 and 16-31 of the chunk) x its 16 columns,
// over D=512 with V_WMMA_F32_16X16X4_F32. Each B-fragment feeds both row tiles.
// A layout: lane = (M = lane&15, half h = lane>>4), per-lane v2f = K (d0+2h, d0+2h+1);
// B mirrors A with N = lane&15 (column k).
__device__ inline void chunk_gemm2(const float* __restrict__ sdir,
                                   const float* __restrict__ cent, int kcol,
                                   v8f& c0, v8f& c1) {
  int lane = threadIdx.x & 31;
  int kn = lane & 15;
  int h  = lane >> 4;
  const float* a0p  = sdir + kn * LDS_STRIDE + 2 * h;
  const float* a1p  = sdir + (16 + kn) * LDS_STRIDE + 2 * h;
  const float* brow = cent + (size_t)(kcol + kn) * DD + 2 * h;
  c0 = (v8f){};
  c1 = (v8f){};
#pragma unroll 4
  for (int d0 = 0; d0 < DD; d0 += 4) {
    v2f bfr = *(const v2f*)(brow + d0);    // L2-resident centroids, reused by 2 wmma
    v2f a0  = *(const v2f*)(a0p + d0);     // ds_load_b64, conflict-free
    v2f a1  = *(const v2f*)(a1p + d0);
    c0 = __builtin_amdgcn_wmma_f32_16x16x4_f32(false, a0, false, bfr, (short)0, c0,
                                               false, false);
    c1 = __builtin_amdgcn_wmma_f32_16x16x4_f32(false, a1, false, bfr, (short)0, c1,
                                               false, false);
  }
}

// ---------------------------------------------------------------------------
// Pass 1: per-(b,k) column softmax stats over L (online max + sumexp).
// Grid (K/128, B), 256 threads = 8 waves, each wave owns 16 columns for all L.
__global__ void k_pass1(const float* __restrict__ x, const float* __restrict__ cent,
                        const float* __restrict__ rscale,
                        float* __restrict__ gM, float* __restrict__ gZ) {
  __shared__ float sdir[LT * LDS_STRIDE];
  __shared__ float srs[DD];
  int b = blockIdx.y;
  int kcol = blockIdx.x * 128 + (threadIdx.x >> 5) * 16;
  int lane = threadIdx.x & 31, kn = lane & 15;

  for (int i = threadIdx.x; i < DD; i += 256) srs[i] = rscale[b * DD + i];

  float run_m = -3.3e38f, run_z = 0.f;
  for (int l0 = 0; l0 < LL; l0 += LT) {
    __syncthreads();
    fill_dir(sdir, srs, x, b, l0);
    __syncthreads();
    v8f c0, c1;
    chunk_gemm2(sdir, cent, kcol, c0, c1);

    // column k = kcol+kn lives in lanes kn and kn+16 (16 rows each after 2 tiles)
    float lm = fmaxf(c0[0], c1[0]);
#pragma unroll
    for (int r = 1; r < 8; ++r) lm = fmaxf(lm, fmaxf(c0[r], c1[r]));
    lm = fmaxf(lm, __shfl_xor(lm, 16));
    float mlog = LAMBDA * lm;
    float s = 0.f;
#pragma unroll
    for (int r = 0; r < 8; ++r) {
      s += __expf(fmaf(LAMBDA, c0[r], -mlog));
      s += __expf(fmaf(LAMBDA, c1[r], -mlog));
    }
    s += __shfl_xor(s, 16);

    float nm = fmaxf(run_m, mlog);
    run_z = run_z * __expf(run_m - nm) + s * __expf(mlog - nm);
    run_m = nm;
  }
  if (lane < 16) {
    gM[b * KK + kcol + kn] = run_m;
    gZ[b * KK + kcol + kn] = run_z;
  }
}

// Pass 2: recompute logits, normalize, accumulate A[b,l] = sum_k a[b,l,k].
__global__ void k_pass2(const float* __restrict__ x, const float* __restrict__ cent,
                        const float* __restrict__ rscale,
                        const float* __restrict__ gM, const float* __restrict__ gZ,
                        float* __restrict__ gA) {
  __shared__ float sdir[LT * LDS_STRIDE];
  __shared__ float srs[DD];
  int b = blockIdx.y;
  int kcol = blockIdx.x * 128 + (threadIdx.x >> 5) * 16;
  int lane = threadIdx.x & 31, kn = lane & 15, h = lane >> 4;

  for (int i = threadIdx.x; i < DD; i += 256) srs[i] = rscale[b * DD + i];
  float m_k  = gM[b * KK + kcol + kn];
  float zinv = 1.0f / gZ[b * KK + kcol + kn];

  for (int l0 = 0; l0 < LL; l0 += LT) {
    __syncthreads();
    fill_dir(sdir, srs, x, b, l0);
    __syncthreads();
    v8f c0, c1;
    chunk_gemm2(sdir, cent, kcol, c0, c1);

    // row sums over the 16 columns held across each 16-lane half
    // tile0 row = 8h + r ; tile1 row = 16 + 8h + r
#pragma unroll
    for (int r = 0; r < 8; ++r) {
      float p = __expf(fmaf(LAMBDA, c0[r], -m_k)) * zinv;
      float q = __expf(fmaf(LAMBDA, c1[r], -m_k)) * zinv;
      p += __shfl_xor(p, 1);  q += __shfl_xor(q, 1);
      p += __shfl_xor(p, 2);  q += __shfl_xor(q, 2);
      p += __shfl_xor(p, 4);  q += __shfl_xor(q, 4);
      p += __shfl_xor(p, 8);  q += __shfl_xor(q, 8);
      if (kn == 0) {
        atomicAdd(&gA[b * LL + l0 + h * 8 + r], p);        // global_atomic_add_f32
        atomicAdd(&gA[b * LL + l0 + 16 + h * 8 + r], q);
      }
    }
  }
}

// out[b,d] = scale/K * ( sum_l A[b,l]*x[b,l,d]*rscale[b,d] - csum[d] )
__global__ void k_final(const float* __restrict__ x, const float* __restrict__ gA,
                        const float* __restrict__ scale, const float* __restrict__ rscale,
                        const float* __restrict__ csum, float* __restrict__ out) {
  int idx = blockIdx.x * blockDim.x + threadIdx.x;
  int b = idx >> 9, d = idx & (DD - 1);
  const float* xp = x + (size_t)b * LL * DD + d;
  const float* ap = gA + b * LL;
  float acc = 0.f;
  for (int l = 0; l < LL; ++l) acc = fmaf(ap[l], xp[(size_t)l * DD], acc);
  out[idx] = scale[idx] * (acc * rscale[idx] - csum[d]) * (1.0f / (float)KK);
}

// ---------------------------------------------------------------------------
extern "C" void kernel_launch(void* const* d_in, const int* in_sizes, int n_in,
                              void* d_out, int out_size, void* d_ws, size_t ws_size,
                              hipStream_t stream) {
  const float* x    = (const float*)d_in[0];   // (B,L,D) fp32
  const float* cent = (const float*)d_in[1];   // (K,D)   fp32

  float* ws     = (float*)d_ws;                // total ~526 KB of scratch
  float* scale  = ws;                          // B*D
  float* rscale = scale  + BB * DD;            // B*D
  float* gM     = rscale + BB * DD;            // B*K
  float* gZ     = gM     + BB * KK;            // B*K
  float* csum   = gZ     + BB * KK;            // D
  float* gA     = csum   + DD;                 // B*L (atomically accumulated)

  hipMemsetAsync(gA, 0, (size_t)BB * LL * sizeof(float), stream);

  k_scale<<<(BB * DD) / 256, 256, 0, stream>>>(x, scale, rscale);
  k_csum<<<(DD + 255) / 256, 256, 0, stream>>>(cent, csum);

  dim3 g(KK / 128, BB);
  k_pass1<<<g, 256, 0, stream>>>(x, cent, rscale, gM, gZ);
  k_pass2<<<g, 256, 0, stream>>>(x, cent, rscale, gM, gZ, gA);

  k_final<<<(BB * DD) / 256, 256, 0, stream>>>(x, gA, scale, rscale, csum,
                                               (float*)d_out);
}